// NonLinearEKV_Conv2d_27144193310964
// MI455X (gfx1250) — compile-verified
//
#include <hip/hip_runtime.h>
#include <cstdint>

// ---------------- problem constants (from reference) ----------------
#define B_   2
#define C_   64
#define H_   32
#define W_   32
#define O_   128
#define OPB  8            // output channels per block (1 per wave, 8 waves)
#define OG_  (O_ / OPB)   // 16 o-groups
#define KK   (C_ * 9)     // 576

#define EKV_ALPHA 0.0005625f
#define EKV_INV   (1.0f / 0.075f)          // 1/(2*n*Vt) = 13.3333
#define EKV_VDI   (0.1f * (1.0f / 0.075f)) // Vd * inv    = 1.3333
#define VSCALE    9.0f

// LDS layout for input rows: region r in {0,1,2} (input rows h-1..h+1),
// line per channel c at 34-DWORD pitch: slot(c,r,w) = r*2176 + c*34 + 1 + w.
// Slots c*34 (left halo) and c*34+33 (right halo) stay zero -> conv padding.
#define LINE_PITCH 34
#define ROW_REGION (C_ * LINE_PITCH)   // 2176
#define SV_DWORDS  (3 * ROW_REGION)    // 6528

typedef unsigned int u32x4 __attribute__((ext_vector_type(4)));
typedef int          i32x4 __attribute__((ext_vector_type(4)));
typedef int          i32x8 __attribute__((ext_vector_type(8)));

#if defined(__gfx1250__) && __has_builtin(__builtin_amdgcn_tensor_load_to_lds)
#define USE_TDM 1
#else
#define USE_TDM 0
#endif

__device__ __forceinline__ float softplus_f(float z) {
    // softplus(z) = max(z,0) + log1p(exp(-|z|)); exact saturation for |z|>~17
    float m = fmaxf(z, 0.0f);
    float e = __expf(-fabsf(z));     // v_exp_f32 path (neg/abs are src modifiers)
    return m + __logf(1.0f + e);     // v_log_f32 path
}

#if USE_TDM
// Build D# (groups 0/1 per CDNA5 ISA 8.3/8.4) and issue TENSOR_LOAD_TO_LDS.
// 2D tile: tile_d0 elements per line, tile_d1 lines, line stride = stride0
// elements. Optional LDS padding: +2 DWORDs after every 32 DWORDs stored.
// This toolchain's builtin takes 6 args (g0 x4, g1 x8, g2 x4, g3 x4, g4 x8, cpol).
__device__ __forceinline__ void tdm_load_2d(uint32_t lds_byte_addr,
                                            const void* gptr,
                                            uint32_t tensor_d0, uint32_t tensor_d1,
                                            uint32_t tile_d0, uint32_t tile_d1,
                                            uint32_t stride0_elems, bool pad) {
    uint64_t ga = (uint64_t)(uintptr_t)gptr;
    u32x4 g0;
    g0[0] = 1u;                                   // count=1, is_restore=0, gather=0
    g0[1] = lds_byte_addr;                        // lds_addr (bytes)
    g0[2] = (uint32_t)ga;                         // global_addr[31:0]
    g0[3] = (uint32_t)(ga >> 32) | (2u << 30);    // global_addr[56:32] | type=2

    uint32_t w0 = (2u << 16);                     // data_size = 4 bytes
    if (pad) w0 |= (1u << 20)                     // pad_enable
              |  (4u << 22)                       // pad_interval: 32 DWORDs
              |  (1u << 25);                      // pad_amount:   2 DWORDs
    i32x8 g1;
    g1[0] = (int)w0;
    g1[1] = (int)(tensor_d0 << 16);                               // dim0 lo16
    g1[2] = (int)((tensor_d0 >> 16) | (tensor_d1 << 16));         // dim0 hi|dim1 lo
    g1[3] = (int)((tensor_d1 >> 16) | (tile_d0 << 16));           // dim1 hi|tile0
    g1[4] = (int)(tile_d1 & 0xFFFFu);                             // tile1, tile2=0
    g1[5] = (int)stride0_elems;                                   // dim0_stride lo32
    g1[6] = 0;
    g1[7] = 0;
    i32x4 z4 = {0, 0, 0, 0};                      // groups 2/3 unused (2D tile)
    i32x8 z8 = {0, 0, 0, 0, 0, 0, 0, 0};          // extra group (unused)
    __builtin_amdgcn_tensor_load_to_lds(g0, g1, z4, z4, z8, 0);
}
#endif

__global__ __launch_bounds__(256) void ekv_conv2d_kernel(
        const float* __restrict__ Vg, const float* __restrict__ Tg,
        float* __restrict__ out) {
    __shared__ float sV[SV_DWORDS];   // 3 halo-padded input rows, all channels
    __shared__ float sT[OPB * KK];    // theta tile (pre-scaled in place)

    const int tid  = threadIdx.x;
    const int lane = tid & 31;
    const int wv   = tid >> 5;                 // wave id 0..7
    const int og   = blockIdx.x % OG_;
    const int h    = (blockIdx.x / OG_) % H_;
    const int b    = blockIdx.x / (OG_ * H_);
    const int o    = og * OPB + wv;

    // Zero the V staging region (halo pads + possibly-missing boundary rows).
    for (int k = tid; k < SV_DWORDS; k += 256) sV[k] = 0.0f;
    __syncthreads();

#if USE_TDM
    if (wv == 0) {  // wave-uniform branch: exactly one wave issues the DMAs
        const uint32_t sv_base = (uint32_t)(uintptr_t)sV;
        const uint32_t st_base = (uint32_t)(uintptr_t)sT;
        // theta tile: contiguous OPB*KK floats, no padding
        tdm_load_2d(st_base, Tg + (size_t)og * OPB * KK,
                    OPB * KK, 1, OPB * KK, 1, OPB * KK, false);
        // input rows h-1..h+1: per row a [32 x 64ch] tile, channel stride H*W,
        // +2 DWORD LDS pad per 32-DWORD line builds the zero halo in place
        #pragma unroll
        for (int r = 0; r < 3; ++r) {
            int hr = h - 1 + r;
            if (hr >= 0 && hr < H_) {
                tdm_load_2d(sv_base + (uint32_t)(r * ROW_REGION + 1) * 4u,
                            Vg + ((size_t)b * C_ * H_ + (size_t)hr) * W_,
                            W_, C_, W_, C_, H_ * W_, true);
            }
        }
        __builtin_amdgcn_s_wait_tensorcnt(0);
    }
#else
    // Fallback: cooperative staged loads (same LDS layout).
    for (int t = tid; t < C_ * 3 * W_; t += 256) {
        int c = t / (3 * W_);
        int rem = t - c * 3 * W_;
        int r = rem / W_;
        int w = rem - r * W_;
        int hr = h - 1 + r;
        if (hr >= 0 && hr < H_)
            sV[r * ROW_REGION + c * LINE_PITCH + 1 + w] =
                Vg[((size_t)(b * C_ + c) * H_ + hr) * W_ + w];
    }
    for (int t = tid; t < OPB * KK; t += 256)
        sT[t] = Tg[(size_t)og * OPB * KK + t];
#endif
    __syncthreads();

    // Pre-transform theta once: theta' = clip(theta, 0, 9) * inv.
    // Saves 2 VALU per tap per lane in the hot loop (amortized 18 ops/thread).
    for (int t = tid; t < OPB * KK; t += 256) {
        float th = sT[t];
        sT[t] = fminf(fmaxf(th, 0.0f), VSCALE) * EKV_INV;
    }
    __syncthreads();

    // ---------------- compute: 576 taps per thread ----------------
    float acc = 0.0f;
    const int tb0 = wv * KK;                    // theta base for this wave's o
    #pragma unroll 2
    for (int c = 0; c < C_; ++c) {
        const int vb = c * LINE_PITCH + lane;   // + r*ROW_REGION + j
        const int tb = tb0 + c * 9;
        #pragma unroll
        for (int i = 0; i < 3; ++i) {
            float v0 = sV[i * ROW_REGION + vb + 0];
            float v1 = sV[i * ROW_REGION + vb + 1];
            float v2 = sV[i * ROW_REGION + vb + 2];
            float vr[3] = {v0, v1, v2};
            #pragma unroll
            for (int j = 0; j < 3; ++j) {
                float thp = sT[tb + i * 3 + j];             // broadcast LDS read
                float z1 = __builtin_fmaf(vr[j], EKV_INV, -thp);
                float z2 = z1 - EKV_VDI;
                float s1 = softplus_f(z1);
                float s2 = softplus_f(z2);
                acc = __builtin_fmaf(s1, s1, acc);
                acc = __builtin_fmaf(-s2, s2, acc);
            }
        }
    }

    out[(((size_t)b * O_ + o) * H_ + h) * W_ + lane] = EKV_ALPHA * acc;
}

extern "C" void kernel_launch(void* const* d_in, const int* in_sizes, int n_in,
                              void* d_out, int out_size, void* d_ws, size_t ws_size,
                              hipStream_t stream) {
    (void)in_sizes; (void)n_in; (void)out_size; (void)d_ws; (void)ws_size;
    const float* V  = (const float*)d_in[0];   // [2,64,32,32] fp32
    const float* Th = (const float*)d_in[1];   // [128,64,3,3] fp32
    float* out = (float*)d_out;                // [2,128,32,32] fp32

    dim3 grid(B_ * H_ * OG_);                  // 1024 blocks
    dim3 block(256);                           // 8 waves (wave32)
    hipLaunchKernelGGL(ekv_conv2d_kernel, grid, block, 0, stream, V, Th, out);
}